// LaplaceMarkovInference_15479062134816
// MI455X (gfx1250) — compile-verified
//
#include <hip/hip_runtime.h>
#include <math.h>

#define BS 256
#define SEQ 16
#define CHUNK (BS * SEQ)       // 4096 elements per block
#define TWO_PI_F 6.2831853071795864f
#define PRECF 1e-6f

typedef float v2f __attribute__((ext_vector_type(2)));
typedef float v8f __attribute__((ext_vector_type(8)));

// ---------------- 2x2 helpers ----------------
struct M2 { float a, b, c, d; };      // [a b; c d]
struct V2 { float x, y; };

__device__ __forceinline__ M2 mm(M2 p, M2 q) {
    return { p.a*q.a + p.b*q.c, p.a*q.b + p.b*q.d,
             p.c*q.a + p.d*q.c, p.c*q.b + p.d*q.d };
}
__device__ __forceinline__ M2 mmT(M2 p, M2 q) {  // p * q^T
    return { p.a*q.a + p.b*q.b, p.a*q.c + p.b*q.d,
             p.c*q.a + p.d*q.b, p.c*q.c + p.d*q.d };
}
__device__ __forceinline__ M2 trn(M2 p) { return { p.a, p.c, p.b, p.d }; }
__device__ __forceinline__ M2 madd(M2 p, M2 q) { return { p.a+q.a, p.b+q.b, p.c+q.c, p.d+q.d }; }
__device__ __forceinline__ M2 inv2(M2 p) {
    float id = 1.0f / (p.a*p.d - p.b*p.c);
    return { p.d*id, -p.b*id, -p.c*id, p.a*id };
}

// ---------------- scan elements ----------------
struct FiltEl { M2 A; V2 b; V2 eta; M2 C; M2 J; };   // 16 floats
struct SmEl   { M2 E; V2 g; M2 L; };                  // 10 floats

__device__ __forceinline__ FiltEl fe_ident() {
    return { {1,0,0,1}, {0,0}, {0,0}, {0,0,0,0}, {0,0,0,0} };
}
__device__ __forceinline__ SmEl sm_ident() {
    return { {1,0,0,1}, {0,0}, {0,0,0,0} };
}

// combine: p earlier in time, q later; result = q o p  (Sarkka & Garcia-Fernandez)
__device__ __forceinline__ FiltEl op_filt(const FiltEl& p, const FiltEl& q) {
    M2 CJ = mm(p.C, q.J);
    M2 Mi = inv2({ 1.f + CJ.a, CJ.b, CJ.c, 1.f + CJ.d });
    M2 A2M = mm(q.A, Mi);
    FiltEl r;
    r.A = mm(A2M, p.A);
    V2 t = { p.b.x + p.C.a*q.eta.x + p.C.b*q.eta.y,
             p.b.y + p.C.c*q.eta.x + p.C.d*q.eta.y };
    r.b = { A2M.a*t.x + A2M.b*t.y + q.b.x,
            A2M.c*t.x + A2M.d*t.y + q.b.y };
    r.C = madd(mmT(mm(A2M, p.C), q.A), q.C);
    M2 W = mm(trn(p.A), trn(Mi));     // (I + J2 C1)^-1 = M^T since C,J symmetric
    V2 u = { q.eta.x - (q.J.a*p.b.x + q.J.b*p.b.y),
             q.eta.y - (q.J.c*p.b.x + q.J.d*p.b.y) };
    r.eta = { W.a*u.x + W.b*u.y + p.eta.x,
              W.c*u.x + W.d*u.y + p.eta.y };
    r.J = madd(mm(mm(W, q.J), p.A), p.J);
    return r;
}

// smoother composition (reverse-time affine maps): result applies q after p
__device__ __forceinline__ SmEl op_sm(const SmEl& p, const SmEl& q) {
    SmEl r;
    r.E = mm(q.E, p.E);
    r.g = { q.E.a*p.g.x + q.E.b*p.g.y + q.g.x,
            q.E.c*p.g.x + q.E.d*p.g.y + q.g.y };
    r.L = madd(q.L, mmT(mm(q.E, p.L), q.E));
    return r;
}

// Matern-3/2 discretization
__device__ __forceinline__ void discretize(float dt, float lam, float p0, float p1,
                                           M2& F, M2& Q) {
    float e = __expf(-lam * dt), ld = lam * dt;
    F = { e*(1.f + ld), e*dt, -e*lam*lam*dt, e*(1.f - ld) };
    float q00 = p0 - (F.a*F.a*p0 + F.b*F.b*p1);
    float q01 =     - (F.a*F.c*p0 + F.b*F.d*p1);
    float q11 = p1 - (F.c*F.c*p0 + F.d*F.d*p1);
    Q = { q00, q01, q01, q11 };
}

// ---------------- elementwise kernels ----------------
__global__ __launch_bounds__(BS) void k_init_f(float* f, int n) {
    int i = blockIdx.x * BS + threadIdx.x;
    if (i < n) f[i] = 0.f;
}

__global__ __launch_bounds__(BS) void k_site(const float* __restrict__ f,
                                             const float* __restrict__ y,
                                             float* nat1, float* nat2,
                                             float* resid, float* Rarr, int n) {
    int i = blockIdx.x * BS + threadIdx.x;
    if (i >= n) return;
    float fv = f[i], yv = y[i];
    float ef = __expf(fv);
    float Lm = fmaxf(ef, PRECF);                 // -h clamped
    float n1 = (yv - ef) + Lm * fv;              // damping == 1
    float n2 = fmaxf(Lm, PRECF);
    nat1[i] = n1; nat2[i] = n2;
    resid[i] = n1 / n2; Rarr[i] = 1.f / n2;
}

// builds scan elements; stages `times` through LDS with CDNA5 async-to-LDS loads
__global__ __launch_bounds__(BS) void k_filt_elems(const float* __restrict__ times,
                                                   const float* __restrict__ resid,
                                                   const float* __restrict__ Rarr,
                                                   const float* lsp, const float* varp,
                                                   FiltEl* __restrict__ E, int n) {
    __shared__ float st[BS];
    int idx = blockIdx.x * BS + threadIdx.x;
    {
        unsigned lds_off = (unsigned)(unsigned long long)&st[threadIdx.x];
        unsigned long long gb = (unsigned long long)times;
        unsigned voff = (unsigned)idx * 4u;
        if (idx < n)
            asm volatile("global_load_async_to_lds_b32 %0, %1, %2"
                         :: "v"(lds_off), "v"(voff), "s"(gb) : "memory");
        asm volatile("s_wait_asynccnt 0x0" ::: "memory");
    }
    __syncthreads();
    if (idx >= n) return;
    float ls = lsp[0], var = varp[0];
    float lam = 1.7320508075688772f / ls;
    float p0 = var, p1 = var * lam * lam;
    float tc = st[threadIdx.x];
    float tp = (threadIdx.x > 0) ? st[threadIdx.x - 1]
                                 : ((idx > 0) ? times[idx - 1] : tc);
    float dt = (idx == 0) ? 0.f : (tc - tp);
    M2 F, Q;
    if (idx == 0) { F = {0,0,0,0}; Q = {p0,0,0,p1}; }   // x0 ~ N(0, Pinf)
    else discretize(dt, lam, p0, p1, F, Q);
    float r = resid[idx], R = Rarr[idx];
    float sv = Q.a + R, iS = 1.f / sv;
    float K0 = Q.a * iS, K1 = Q.c * iS;
    FiltEl el;
    el.A = { (1.f - K0)*F.a, (1.f - K0)*F.b, F.c - K1*F.a, F.d - K1*F.b };
    el.b = { K0 * r, K1 * r };
    el.C = { Q.a - K0*Q.a, Q.b - K0*Q.b, Q.c - K1*Q.a, Q.d - K1*Q.b };
    el.eta = { F.a * r * iS, F.b * r * iS };
    el.J = { F.a*F.a*iS, F.a*F.b*iS, F.a*F.b*iS, F.b*F.b*iS };
    E[idx] = el;
}

// ---------------- filter scan (3 phases) ----------------
__global__ __launch_bounds__(BS) void k_scanA_f(const FiltEl* __restrict__ E,
                                                FiltEl* __restrict__ agg, int n) {
    __shared__ FiltEl lds[BS];
    int t = threadIdx.x;
    int base = blockIdx.x * CHUNK + t * SEQ;
    FiltEl a = fe_ident();
    for (int i = 0; i < SEQ; ++i) { int k = base + i; if (k < n) a = op_filt(a, E[k]); }
    lds[t] = a; __syncthreads();
    for (int off = 1; off < BS; off <<= 1) {
        FiltEl oth; bool h = t >= off;
        if (h) oth = lds[t - off];
        __syncthreads();
        if (h) a = op_filt(oth, a);
        lds[t] = a; __syncthreads();
    }
    if (t == BS - 1) agg[blockIdx.x] = a;
}

__global__ __launch_bounds__(BS) void k_scanB_f(const FiltEl* __restrict__ agg,
                                                FiltEl* __restrict__ excl, int nblk) {
    __shared__ FiltEl lds[BS];
    int t = threadIdx.x;
    FiltEl a = (t < nblk) ? agg[t] : fe_ident();
    lds[t] = a; __syncthreads();
    for (int off = 1; off < BS; off <<= 1) {
        FiltEl oth; bool h = t >= off;
        if (h) oth = lds[t - off];
        __syncthreads();
        if (h) a = op_filt(oth, a);
        lds[t] = a; __syncthreads();
    }
    if (t < nblk) excl[t] = (t == 0) ? fe_ident() : lds[t - 1];
}

__global__ __launch_bounds__(BS) void k_scanC_f(const FiltEl* __restrict__ E,
                                                const FiltEl* __restrict__ excl,
                                                float2* __restrict__ mf,
                                                float4* __restrict__ pf, int n) {
    __shared__ FiltEl lds[BS];
    int t = threadIdx.x;
    int base = blockIdx.x * CHUNK + t * SEQ;
    FiltEl a = fe_ident();
    for (int i = 0; i < SEQ; ++i) { int k = base + i; if (k < n) a = op_filt(a, E[k]); }
    lds[t] = a; __syncthreads();
    for (int off = 1; off < BS; off <<= 1) {
        FiltEl oth; bool h = t >= off;
        if (h) oth = lds[t - off];
        __syncthreads();
        if (h) a = op_filt(oth, a);
        lds[t] = a; __syncthreads();
    }
    FiltEl carry = excl[blockIdx.x];
    if (t > 0) carry = op_filt(carry, lds[t - 1]);
    for (int i = 0; i < SEQ; ++i) {
        int k = base + i; if (k >= n) break;
        __builtin_prefetch(&E[k + 4], 0, 1);
        carry = op_filt(carry, E[k]);
        mf[k] = make_float2(carry.b.x, carry.b.y);       // A-prefix == 0 => exact marginal
        pf[k] = make_float4(carry.C.a, carry.C.b, carry.C.d, 0.f);
    }
}

// predicted moments + per-step filter log-lik terms
__global__ __launch_bounds__(BS) void k_pred(const float* __restrict__ times,
                                             const float* lsp, const float* varp,
                                             const float2* __restrict__ mf,
                                             const float4* __restrict__ pf,
                                             const float* __restrict__ resid,
                                             const float* __restrict__ Rarr,
                                             float2* __restrict__ mp,
                                             float4* __restrict__ pp,
                                             float* __restrict__ llf, int n) {
    int k = blockIdx.x * BS + threadIdx.x;
    if (k >= n) return;
    float ls = lsp[0], var = varp[0];
    float lam = 1.7320508075688772f / ls;
    float p0 = var, p1 = var * lam * lam;
    float mpx, s;
    if (k == 0) {
        mp[0] = make_float2(0.f, 0.f);
        pp[0] = make_float4(p0, 0.f, p1, 0.f);
        mpx = 0.f; s = p0 + Rarr[0];
    } else {
        float dt = times[k] - times[k - 1];
        M2 F, Q; discretize(dt, lam, p0, p1, F, Q);
        float2 m1 = mf[k - 1]; float4 P1 = pf[k - 1];
        M2 P = { P1.x, P1.y, P1.y, P1.z };
        float mx = F.a*m1.x + F.b*m1.y, my = F.c*m1.x + F.d*m1.y;
        M2 PP = madd(mmT(mm(F, P), F), Q);
        mp[k] = make_float2(mx, my);
        pp[k] = make_float4(PP.a, 0.5f*(PP.b + PP.c), PP.d, 0.f);
        mpx = mx; s = PP.a + Rarr[k];
    }
    float v = resid[k] - mpx;
    llf[k] = -0.5f * (logf(TWO_PI_F * s) + v * v / s);
}

// ---------------- smoother elements + reverse scan ----------------
__global__ __launch_bounds__(BS) void k_sm_elems(const float* __restrict__ times,
                                                 const float* lsp, const float* varp,
                                                 const float2* __restrict__ mf,
                                                 const float4* __restrict__ pf,
                                                 const float2* __restrict__ mp,
                                                 const float4* __restrict__ pp,
                                                 SmEl* __restrict__ Sm,
                                                 float* __restrict__ fout,
                                                 float* __restrict__ varout, int n) {
    int k = blockIdx.x * BS + threadIdx.x;
    if (k >= n) return;
    if (k == n - 1) {                 // terminal value; pad scan slot j=n-1 with identity
        float2 m = mf[k]; float4 P = pf[k];
        fout[k] = m.x;
        varout[k] = fmaxf(P.x, 1e-12f);
        Sm[n - 1] = sm_ident();
        return;
    }
    int j = n - 2 - k;
    float ls = lsp[0], var = varp[0];
    float lam = 1.7320508075688772f / ls;
    float p0 = var, p1 = var * lam * lam;
    float dt = times[k + 1] - times[k];
    M2 A1, Qd; discretize(dt, lam, p0, p1, A1, Qd);
    float2 m = mf[k]; float4 P = pf[k];
    float2 m1 = mp[k + 1]; float4 P1 = pp[k + 1];
    M2 Pf = { P.x, P.y, P.y, P.z };
    M2 Pp1 = { P1.x, P1.y, P1.y, P1.z };
    M2 G = mm(mmT(Pf, A1), inv2(Pp1));           // Pf A1^T Pp1^-1
    M2 GPG = mmT(mm(G, Pp1), G);
    SmEl e;
    e.E = G;
    e.g = { m.x - (G.a*m1.x + G.b*m1.y), m.y - (G.c*m1.x + G.d*m1.y) };
    e.L = { Pf.a - GPG.a, Pf.b - GPG.b, Pf.c - GPG.c, Pf.d - GPG.d };
    Sm[j] = e;
}

__global__ __launch_bounds__(BS) void k_scanA_s(const SmEl* __restrict__ Sm,
                                                SmEl* __restrict__ agg, int n) {
    __shared__ SmEl lds[BS];
    int t = threadIdx.x;
    int base = blockIdx.x * CHUNK + t * SEQ;
    SmEl a = sm_ident();
    for (int i = 0; i < SEQ; ++i) { int k = base + i; if (k < n) a = op_sm(a, Sm[k]); }
    lds[t] = a; __syncthreads();
    for (int off = 1; off < BS; off <<= 1) {
        SmEl oth; bool h = t >= off;
        if (h) oth = lds[t - off];
        __syncthreads();
        if (h) a = op_sm(oth, a);
        lds[t] = a; __syncthreads();
    }
    if (t == BS - 1) agg[blockIdx.x] = a;
}

__global__ __launch_bounds__(BS) void k_scanB_s(const SmEl* __restrict__ agg,
                                                SmEl* __restrict__ excl, int nblk) {
    __shared__ SmEl lds[BS];
    int t = threadIdx.x;
    SmEl a = (t < nblk) ? agg[t] : sm_ident();
    lds[t] = a; __syncthreads();
    for (int off = 1; off < BS; off <<= 1) {
        SmEl oth; bool h = t >= off;
        if (h) oth = lds[t - off];
        __syncthreads();
        if (h) a = op_sm(oth, a);
        lds[t] = a; __syncthreads();
    }
    if (t < nblk) excl[t] = (t == 0) ? sm_ident() : lds[t - 1];
}

__global__ __launch_bounds__(BS) void k_scanC_s(const SmEl* __restrict__ Sm,
                                                const SmEl* __restrict__ excl,
                                                const float2* __restrict__ mf,
                                                const float4* __restrict__ pf,
                                                float* __restrict__ fout,
                                                float* __restrict__ varout, int n) {
    __shared__ SmEl lds[BS];
    int t = threadIdx.x;
    int base = blockIdx.x * CHUNK + t * SEQ;
    float2 mt = mf[n - 1]; float4 pt = pf[n - 1];
    SmEl a = sm_ident();
    for (int i = 0; i < SEQ; ++i) { int k = base + i; if (k < n) a = op_sm(a, Sm[k]); }
    lds[t] = a; __syncthreads();
    for (int off = 1; off < BS; off <<= 1) {
        SmEl oth; bool h = t >= off;
        if (h) oth = lds[t - off];
        __syncthreads();
        if (h) a = op_sm(oth, a);
        lds[t] = a; __syncthreads();
    }
    SmEl carry = excl[blockIdx.x];
    if (t > 0) carry = op_sm(carry, lds[t - 1]);
    for (int i = 0; i < SEQ; ++i) {
        int j = base + i; if (j >= n) break;
        __builtin_prefetch(&Sm[j + 4], 0, 1);
        carry = op_sm(carry, Sm[j]);
        if (j <= n - 2) {
            int k = n - 2 - j;
            float fm = carry.E.a*mt.x + carry.E.b*mt.y + carry.g.x;
            float e0 = carry.E.a, e1 = carry.E.b;
            float pv = carry.L.a + e0*(pt.x*e0 + pt.y*e1) + e1*(pt.y*e0 + pt.z*e1);
            fout[k] = fm;
            varout[k] = fmaxf(pv, 1e-12f);
        }
    }
}

// ---------------- final log-marginal terms + WMMA reduction ----------------
__global__ __launch_bounds__(BS) void k_llterms(const float* __restrict__ qm,
                                                const float* __restrict__ y,
                                                const float* __restrict__ nat1,
                                                const float* __restrict__ nat2,
                                                const float* __restrict__ llf,
                                                float* __restrict__ llt, int n) {
    int i = blockIdx.x * BS + threadIdx.x;
    if (i >= n) return;
    float q = qm[i], yv = y[i], n1 = nat1[i], n2 = nat2[i];
    float lld = yv * q - __expf(q) - lgammaf(yv + 1.f);
    float tgt = n1 / n2;
    float llp = -0.5f * logf(TWO_PI_F / n2) - 0.5f * n2 * (q - tgt) * (q - tgt);
    llt[i] = llf[i] + lld - llp;
}

// sum reduction using V_WMMA_F32_16X16X4_F32: D = A(16x4)*ones(4x16) + D.
// Main loop: one unconditional b64 load + one WMMA per 64 elements (EXEC all-ones,
// no per-element bounds branching). All 16 columns of D are identical; the column
// total is recovered with one shfl_xor(16). Scalar tail handled by lane 0 of block 0
// AFTER all WMMAs so the EXEC==all-1s WMMA requirement is never violated.
__global__ __launch_bounds__(32) void k_reduce_wmma(const float* __restrict__ src,
                                                    int nn, float* __restrict__ dst) {
    v8f acc = { 0.f, 0.f, 0.f, 0.f, 0.f, 0.f, 0.f, 0.f };
    v2f ones; ones.x = 1.f; ones.y = 1.f;
    const int t = threadIdx.x;
    const int nfull = nn & ~63;                       // multiple of 64
    const int stride = gridDim.x * 64;
    const float2* s2 = (const float2*)src;
    for (int base = blockIdx.x * 64; base < nfull; base += stride) {
        __builtin_prefetch(&src[base + stride], 0, 1);
        float2 q = s2[(base >> 1) + t];               // coalesced b64 load
        v2f a; a.x = q.x; a.y = q.y;
        acc = __builtin_amdgcn_wmma_f32_16x16x4_f32(false, a, false, ones,
                                                    (short)0, acc, false, false);
    }
    float s = 0.f;
    for (int r = 0; r < 8; ++r) s += acc[r];
    s += __shfl_xor(s, 16, 32);       // rows 0-7 + rows 8-15 of one column = total
    if (blockIdx.x == 0 && t == 0)
        for (int i = nfull; i < nn; ++i) s += src[i]; // generic tail
    if (t == 0) dst[blockIdx.x] = s;
}

// ---------------- host orchestration ----------------
extern "C" void kernel_launch(void* const* d_in, const int* in_sizes, int n_in,
                              void* d_out, int out_size, void* d_ws, size_t ws_size,
                              hipStream_t stream) {
    const float* times = (const float*)d_in[0];
    const float* y     = (const float*)d_in[1];
    const float* lsp   = (const float*)d_in[2];
    const float* varp  = (const float*)d_in[3];
    const int n = in_sizes[0];

    float* out_mean = (float*)d_out;
    float* out_var  = out_mean + n;
    float* out_lm   = out_mean + 2 * n;

    float* w = (float*)d_ws;
    size_t N = (size_t)n;
    float*  f_    = w;
    float*  nat1  = w + 1 * N;
    float*  nat2  = w + 2 * N;
    float*  resid = w + 3 * N;
    float*  Rarr  = w + 4 * N;
    float*  vtmp  = w + 5 * N;
    float*  llf   = w + 6 * N;
    float*  llt   = w + 7 * N;
    float2* mf    = (float2*)(w + 8 * N);
    float4* pf    = (float4*)(w + 10 * N);
    float2* mp    = (float2*)(w + 14 * N);
    float4* pp    = (float4*)(w + 16 * N);
    FiltEl* E     = (FiltEl*)(w + 20 * N);   // 16N floats
    SmEl*   Sm    = (SmEl*)  (w + 36 * N);   // 10N floats
    float*  tail  = w + 46 * N;
    FiltEl* aggF  = (FiltEl*)(tail);
    FiltEl* exF   = (FiltEl*)(tail + 4096);
    SmEl*   aggS  = (SmEl*)  (tail + 8192);
    SmEl*   exS   = (SmEl*)  (tail + 8192 + 2560);
    float*  part  = tail + 8192 + 5120;

    const int eb   = (n + BS - 1) / BS;
    const int nblk = (n + CHUNK - 1) / CHUNK;

    k_init_f<<<eb, BS, 0, stream>>>(f_, n);

    for (int it = 0; it < 6; ++it) {
        const bool fin = (it == 5);
        float* fo = fin ? out_mean : f_;
        float* vo = fin ? out_var  : vtmp;

        k_site<<<eb, BS, 0, stream>>>(f_, y, nat1, nat2, resid, Rarr, n);
        k_filt_elems<<<eb, BS, 0, stream>>>(times, resid, Rarr, lsp, varp, E, n);
        k_scanA_f<<<nblk, BS, 0, stream>>>(E, aggF, n);
        k_scanB_f<<<1, BS, 0, stream>>>(aggF, exF, nblk);
        k_scanC_f<<<nblk, BS, 0, stream>>>(E, exF, mf, pf, n);
        k_pred<<<eb, BS, 0, stream>>>(times, lsp, varp, mf, pf, resid, Rarr, mp, pp, llf, n);
        k_sm_elems<<<eb, BS, 0, stream>>>(times, lsp, varp, mf, pf, mp, pp, Sm, fo, vo, n);
        k_scanA_s<<<nblk, BS, 0, stream>>>(Sm, aggS, n);
        k_scanB_s<<<1, BS, 0, stream>>>(aggS, exS, nblk);
        k_scanC_s<<<nblk, BS, 0, stream>>>(Sm, exS, mf, pf, fo, vo, n);
    }

    // log_marg + ll_data - ll_pseudo, summed with the WMMA reduction engine
    k_llterms<<<eb, BS, 0, stream>>>(out_mean, y, nat1, nat2, llf, llt, n);
    k_reduce_wmma<<<64, 32, 0, stream>>>(llt, n, part);
    k_reduce_wmma<<<1, 32, 0, stream>>>(part, 64, out_lm);
}